// OutputLayer_1142461301029
// MI455X (gfx1250) — compile-verified
//
#include <hip/hip_runtime.h>

typedef __attribute__((ext_vector_type(2))) float v2f;
typedef __attribute__((ext_vector_type(4))) float v4f;
typedef __attribute__((ext_vector_type(8))) float v8f;

#define D_DIM 512
#define S_DIM 2048

// ---------------------------------------------------------------------------
// Kernel 1: y[row] = dot(x[row, 0:512], W) + b   for row in [0, B*S)
//
// Wave32: each wave computes 16 rows with V_WMMA_F32_16X16X4_F32.
//   A (16x4 f32): lane L (m = L%16) holds x[row, k + koff], x[row, k + koff+1]
//                 where koff = 2*(L>=16)             -> per-lane b64 loads
//   B (4x16 f32): B[k, n] = W[k] for all n: lanes 0-15 need W[k],W[k+1],
//                 lanes 16-31 need W[k+2],W[k+3]     -> ds_load_b64 from LDS
//   C (16x16 f32): y_m replicated across all 16 columns; lanes 0 and 16 own
//                 columns holding m = 0..7 and 8..15 in VGPRs 0..7.
// Two independent accumulator chains (K halves) for WMMA latency hiding.
// Block = 256 threads = 8 waves = 128 rows.
// ---------------------------------------------------------------------------
__global__ __launch_bounds__(256) void gemv_wmma_f32(
    const float* __restrict__ x,
    const float* __restrict__ W,
    const float* __restrict__ bias,
    float* __restrict__ y)
{
    __shared__ float wlds[D_DIM];
    for (int i = threadIdx.x; i < D_DIM; i += 256) wlds[i] = W[i];
    __syncthreads();

    const int lane    = threadIdx.x & 31;
    const int wave    = threadIdx.x >> 5;
    const int rowBase = blockIdx.x * 128 + wave * 16;
    const int m       = lane & 15;
    const int koff    = (lane >> 4) << 1;   // 0 for lanes 0-15, 2 for 16-31

    const float* xp = x + (size_t)(rowBase + m) * D_DIM + koff;
    const float* wp = wlds + koff;

    const float bv = bias[0];
    v8f c0 = {bv, bv, bv, bv, bv, bv, bv, bv};   // bias folded into C init
    v8f c1 = {0.f, 0.f, 0.f, 0.f, 0.f, 0.f, 0.f, 0.f};

#pragma unroll 8
    for (int k = 0; k < D_DIM / 2; k += 4) {
        const v2f a0  = *reinterpret_cast<const v2f*>(xp + k);              // global b64
        const v2f b0  = *reinterpret_cast<const v2f*>(wp + k);              // LDS b64
        const v2f a1  = *reinterpret_cast<const v2f*>(xp + k + D_DIM / 2);  // global b64
        const v2f b1  = *reinterpret_cast<const v2f*>(wp + k + D_DIM / 2);  // LDS b64
        // 8 args: (neg_a, A, neg_b, B, c_mod, C, reuse_a, reuse_b)
        c0 = __builtin_amdgcn_wmma_f32_16x16x4_f32(
            false, a0, false, b0, (short)0, c0, false, false);
        c1 = __builtin_amdgcn_wmma_f32_16x16x4_f32(
            false, a1, false, b1, (short)0, c1, false, false);
    }

    if ((lane & 15) == 0) {
        float* o = y + rowBase + ((lane >> 4) << 3);  // lane0 -> m 0..7, lane16 -> m 8..15
        v4f lo = {c0[0] + c1[0], c0[1] + c1[1], c0[2] + c1[2], c0[3] + c1[3]};
        v4f hi = {c0[4] + c1[4], c0[5] + c1[5], c0[6] + c1[6], c0[7] + c1[7]};
        *reinterpret_cast<v4f*>(o)     = lo;   // global_store_b128
        *reinterpret_cast<v4f*>(o + 4) = hi;   // global_store_b128
    }
}

// ---------------------------------------------------------------------------
// Kernel 2: per batch b (one 1024-thread block, 2 elements/thread):
//   mn/mx over entries with mask==0, normalize those, keep masked raw,
//   stable-partition (unmasked first) via prefix sum of valid flags.
// Operates in place on y/out: each block reads only its own row, and all
// reads complete (values consumed into LDS) before the first barrier.
// ---------------------------------------------------------------------------
__global__ __launch_bounds__(1024) void norm_reorder(
    float* __restrict__ yio,
    const int* __restrict__ mask)
{
    const int b = blockIdx.x;
    const int t = threadIdx.x;              // 0..1023
    const int s0 = 2 * t, s1 = 2 * t + 1;
    const size_t base = (size_t)b * S_DIM;

    __shared__ float rmin[1024];
    __shared__ float rmax[1024];
    __shared__ int   scanb[1024];

    const float y0 = yio[base + s0];
    const float y1 = yio[base + s1];
    const bool  v0 = (mask[base + s0] == 0);
    const bool  v1 = (mask[base + s1] == 0);

    const float INF = __builtin_inff();
    rmin[t] = fminf(v0 ? y0 :  INF, v1 ? y1 :  INF);
    rmax[t] = fmaxf(v0 ? y0 : -INF, v1 ? y1 : -INF);
    const int cnt = (int)v0 + (int)v1;
    scanb[t] = cnt;
    __syncthreads();

    // min/max tree reduction
    for (int off = 512; off > 0; off >>= 1) {
        if (t < off) {
            rmin[t] = fminf(rmin[t], rmin[t + off]);
            rmax[t] = fmaxf(rmax[t], rmax[t + off]);
        }
        __syncthreads();
    }
    const float mn = rmin[0];
    const float mx = rmax[0];

    // Hillis-Steele inclusive scan of valid counts (read-barrier-write-barrier)
    for (int off = 1; off < 1024; off <<= 1) {
        const int mine = scanb[t];
        const int add  = (t >= off) ? scanb[t - off] : 0;
        __syncthreads();
        scanb[t] = mine + add;
        __syncthreads();
    }
    const int incl  = scanb[t];
    const int excl  = incl - cnt;          // #valid in [0, s0)
    const int total = scanb[1023];         // total valid in row

    const int pv0 = excl;
    const int pv1 = excl + (int)v0;
    const int pos0 = v0 ? pv0 : (total + (s0 - pv0));
    const int pos1 = v1 ? pv1 : (total + (s1 - pv1));

    const float d = mx - mn;
    const float o0 = v0 ? ((y0 - mn) / d) : y0;
    const float o1 = v1 ? ((y1 - mn) / d) : y1;

    yio[base + pos0] = o0;
    yio[base + pos1] = o1;
}

extern "C" void kernel_launch(void* const* d_in, const int* in_sizes, int n_in,
                              void* d_out, int out_size, void* d_ws, size_t ws_size,
                              hipStream_t stream)
{
    const float* x    = (const float*)d_in[0];   // [B, S, D] f32
    const int*   mask = (const int*)  d_in[1];   // [B, S] i32
    const float* W    = (const float*)d_in[2];   // [1, D] f32
    const float* bias = (const float*)d_in[3];   // [1] f32
    float* out = (float*)d_out;                  // [B, S] f32

    const int rows = in_sizes[1];                // B * S
    const int B    = rows / S_DIM;

    // Stage 1: GEMV via f32 WMMA, writes y into d_out.
    gemv_wmma_f32<<<rows / 128, 256, 0, stream>>>(x, W, bias, out);
    // Stage 2: in-place per-batch normalize + stable partition.
    norm_reorder<<<B, 1024, 0, stream>>>(out, mask);

    (void)n_in; (void)out_size; (void)d_ws; (void)ws_size;
}